// Attention_77824807403911
// MI455X (gfx1250) — compile-verified
//
#include <hip/hip_runtime.h>

typedef __attribute__((ext_vector_type(2))) float v2f;
typedef __attribute__((ext_vector_type(8))) float v8f;
typedef __attribute__((ext_vector_type(4))) int   v4i;

// D = A(16x4) * B(4x16) + C, fp32, wave32.
#define WMMA_F32(A, B, C) \
  __builtin_amdgcn_wmma_f32_16x16x4_f32(false, (A), false, (B), (short)0, (C), false, false)

#if __has_builtin(__builtin_amdgcn_global_load_async_to_lds_b128) && \
    __has_builtin(__builtin_amdgcn_s_wait_asynccnt)
#define USE_ASYNC_LDS 1
typedef __attribute__((address_space(1))) v4i as1_v4i;  // global
typedef __attribute__((address_space(3))) v4i as3_v4i;  // LDS
#endif

constexpr int Bb = 8;
constexpr int Ss = 1024;
constexpr int Dd = 256;
constexpr int Hh = 16;
constexpr int HD = 16;
constexpr int Mrows = Bb * Ss;  // 8192

// ---------------------------------------------------------------------------
// Kernel 1: q = x@wq^T + bq ; v = x@wv^T + bv ; RoPE both; write [b,h,s,hd].
// Block = 128 threads (4 waves). Block owns 16 M-rows; wave w owns N cols
// [64w, 64w+64). A fragment (from x) shared between the Q and V GEMMs.
// B fragments hoisted into distinct registers per k-step so the loads clause
// up instead of ping-ponging on s_wait_loadcnt 0.
// ---------------------------------------------------------------------------
__global__ __launch_bounds__(128) void qv_proj_rope_kernel(
    const float* __restrict__ x,
    const float* __restrict__ wq_w, const float* __restrict__ wq_b,
    const float* __restrict__ wv_w, const float* __restrict__ wv_b,
    float* __restrict__ xq, float* __restrict__ xv)
{
  const int lane = threadIdx.x & 31;
  const int wid  = threadIdx.x >> 5;
  const int m0   = blockIdx.x * 16;
  const int nb   = wid * 64;
  const int l16  = lane & 15;
  const int khi  = (lane >= 16) ? 2 : 0;   // A/B layout: hi half-wave holds K+2
  const int row  = m0 + l16;

  v8f accQ[4], accV[4];
#pragma unroll
  for (int t = 0; t < 4; ++t) { accQ[t] = {}; accV[t] = {}; }

  const float* xrow = x + (size_t)row * Dd;
#pragma unroll 2
  for (int k0 = 0; k0 < Dd; k0 += 4) {
    v2f a;
    a.x = xrow[k0 + khi];
    a.y = xrow[k0 + khi + 1];
    __builtin_prefetch(xrow + k0 + 64, 0, 3);
    v2f bq[4], bv[4];
#pragma unroll
    for (int t = 0; t < 4; ++t) {
      const int n = nb + t * 16 + l16;
      const float* wq = wq_w + (size_t)n * Dd + k0 + khi;  // B[k][n] = w[n][k]
      const float* wv = wv_w + (size_t)n * Dd + k0 + khi;
      bq[t].x = wq[0]; bq[t].y = wq[1];
      bv[t].x = wv[0]; bv[t].y = wv[1];
    }
#pragma unroll
    for (int t = 0; t < 4; ++t) {
      accQ[t] = WMMA_F32(a, bq[t], accQ[t]);
      accV[t] = WMMA_F32(a, bv[t], accV[t]);
    }
  }

  // Bias + RoPE + scatter to [b,h,s,hd].
  // C layout: VGPR e holds M = e (lanes 0-15) / e+8 (lanes 16-31), N = lane%16.
  // RoPE pair (2p, 2p+1) lives in adjacent lanes -> shfl_xor(1).
#pragma unroll
  for (int t = 0; t < 4; ++t) {
    const int n     = nb + t * 16 + l16;
    const float bq  = wq_b[n];
    const float bv  = wv_b[n];
    const float th  = (n < 128) ? 1.0f : 1.0e-4f;  // theta^-((2i)//d), d=256
    const bool  odd = (n & 1) != 0;
    const int   h   = n >> 4;
    const int   hd  = n & 15;
#pragma unroll
    for (int e = 0; e < 8; ++e) {
      const int mr = m0 + e + ((lane >= 16) ? 8 : 0);
      const int s  = mr & (Ss - 1);
      const int bi = mr >> 10;
      const float ang = (float)(s + 1) * th;      // 1-indexed positions
      float sn, cs;
      __sincosf(ang, &sn, &cs);
      const size_t o = (((size_t)bi * Hh + h) * Ss + s) * HD + hd;

      float q  = accQ[t][e] + bq;
      float qp = __shfl_xor(q, 1, 32);
      float qe = odd ? qp : q;
      float qo = odd ? q : qp;
      xq[o] = odd ? (qe * cs + qo * sn) : (qe * sn - qo * cs);

      float v  = accV[t][e] + bv;
      float vp = __shfl_xor(v, 1, 32);
      float ve = odd ? vp : v;
      float vo = odd ? v : vp;
      xv[o] = odd ? (ve * cs + vo * sn) : (ve * sn - vo * cs);
    }
  }
}

// ---------------------------------------------------------------------------
// Kernel 2: per (b,h): scoresT = Xq_k . (0.25*Xq_q)^T ; attT += V^T . scoresT.
// Block = 512 threads (16 waves), one block per (b,h). Keys/values staged in
// LDS in chunks of 128 rows via gfx1250 async-to-LDS b128 copies (one op per
// thread per buffer). Each wave owns 4 q-tiles. 1/sqrt(HD) folded into the
// resident Q fragments so staging is a pure byte copy.
// ---------------------------------------------------------------------------
__global__ __launch_bounds__(512) void attn_kernel(
    const float* __restrict__ xq, const float* __restrict__ xv,
    float* __restrict__ att)
{
  __shared__ float sK[128 * HD];
  __shared__ float sV[128 * HD];

  const int lane = threadIdx.x & 31;
  const int wid  = threadIdx.x >> 5;
  const int l16  = lane & 15;
  const int khi  = (lane >= 16) ? 2 : 0;
  const bool lo  = (lane < 16);
  const int bh   = blockIdx.x;   // b*H + h
  const int b    = bh >> 4;
  const int h    = bh & 15;

  const float* q_g = xq + (size_t)bh * Ss * HD;
  const float* v_g = xv + (size_t)bh * Ss * HD;

  // Resident B-fragments of this wave's 4 Q tiles: B[kk=hd][n=qpos], x 0.25.
  v2f qb[4][4];
#pragma unroll
  for (int i = 0; i < 4; ++i) {
    const int q0 = (wid * 4 + i) * 16;
    const float* qr = q_g + (size_t)(q0 + l16) * HD;
#pragma unroll
    for (int c = 0; c < 4; ++c) {
      qb[i][c].x = 0.25f * qr[4 * c + khi];
      qb[i][c].y = 0.25f * qr[4 * c + khi + 1];
    }
  }

  v8f acc[4];  // attT accumulators (M=hd, N=qpos)
#pragma unroll
  for (int i = 0; i < 4; ++i) acc[i] = {};

  for (int cc = 0; cc < 8; ++cc) {         // 8 chunks x 128 key rows
    __syncthreads();
    const size_t gbase = (size_t)cc * 128 * HD;
#if defined(USE_ASYNC_LDS)
    {
      const int fo = threadIdx.x * 4;      // 4 floats = 16 bytes per thread
      __builtin_amdgcn_global_load_async_to_lds_b128(
          (as1_v4i*)(q_g + gbase + fo), (as3_v4i*)(sK + fo), 0, 0);
      __builtin_amdgcn_global_load_async_to_lds_b128(
          (as1_v4i*)(v_g + gbase + fo), (as3_v4i*)(sV + fo), 0, 0);
      __builtin_amdgcn_s_wait_asynccnt(0);
    }
#else
#pragma unroll
    for (int t = 0; t < 4; ++t) {
      const int idx = threadIdx.x + t * 512;
      sK[idx] = q_g[gbase + idx];
      sV[idx] = v_g[gbase + idx];
    }
#endif
    __syncthreads();

    for (int j = 0; j < 8; ++j) {          // k-tile inside chunk
      // A-fragments of the key tile and of V^T: invariant across q-tiles.
      v2f aK[4], vA[4];
#pragma unroll
      for (int c = 0; c < 4; ++c) {
        aK[c].x = sK[(j * 16 + l16) * HD + 4 * c + khi];
        aK[c].y = sK[(j * 16 + l16) * HD + 4 * c + khi + 1];
        vA[c].x = sV[(j * 16 + 4 * c + khi) * HD + l16];
        vA[c].y = sV[(j * 16 + 4 * c + khi + 1) * HD + l16];
      }
#pragma unroll
      for (int i = 0; i < 4; ++i) {        // this wave's q-tiles
        // --- GEMM 1: scoresT(16x16) = K_tile(16xHD) x (0.25*Q_tile)^T ---
        v8f sT = {};
#pragma unroll
        for (int c = 0; c < 4; ++c) sT = WMMA_F32(aK[c], qb[i][c], sT);
        // --- GEMM 2: attT += V^T(HDx16key) x scoresT(16keyx16q) ---
        // Convert sT (C layout) to B fragments via half-wave exchanges:
        // the C layout splits M(=key) at 8 between half-waves, so
        // shfl_xor(16) swaps exactly the elements the other half needs.
#pragma unroll
        for (int g = 0; g < 2; ++g) {
          const float x0 = __shfl_xor(sT[4 * g + 0], 16, 32);
          const float x1 = __shfl_xor(sT[4 * g + 1], 16, 32);
          const float x2 = __shfl_xor(sT[4 * g + 2], 16, 32);
          const float x3 = __shfl_xor(sT[4 * g + 3], 16, 32);
          {
            v2f bb;                        // keys 4g .. 4g+3  (c = g)
            bb.x = lo ? sT[4 * g + 0] : x2;
            bb.y = lo ? sT[4 * g + 1] : x3;
            acc[i] = WMMA_F32(vA[g], bb, acc[i]);
          }
          {
            v2f bb;                        // keys 4g+8 .. 4g+11  (c = g+2)
            bb.x = lo ? x0 : sT[4 * g + 2];
            bb.y = lo ? x1 : sT[4 * g + 3];
            acc[i] = WMMA_F32(vA[g + 2], bb, acc[i]);
          }
        }
      }
    }
  }

  // Store attT: value (M=hd, N=q) -> att[b, s=q, h*16+hd]  ([b,s,d] layout)
#pragma unroll
  for (int i = 0; i < 4; ++i) {
    const int s = (wid * 4 + i) * 16 + l16;
#pragma unroll
    for (int e = 0; e < 8; ++e) {
      const int hd = e + ((lane >= 16) ? 8 : 0);
      att[((size_t)b * Ss + s) * Dd + h * HD + hd] = acc[i][e];
    }
  }
}

// ---------------------------------------------------------------------------
// Kernel 3: out = att @ wo^T + wo_b.  Same tiling as kernel 1.
// ---------------------------------------------------------------------------
__global__ __launch_bounds__(128) void out_proj_kernel(
    const float* __restrict__ att,
    const float* __restrict__ wo_w, const float* __restrict__ wo_b,
    float* __restrict__ out)
{
  const int lane = threadIdx.x & 31;
  const int wid  = threadIdx.x >> 5;
  const int m0   = blockIdx.x * 16;
  const int nb   = wid * 64;
  const int l16  = lane & 15;
  const int khi  = (lane >= 16) ? 2 : 0;
  const int row  = m0 + l16;

  v8f acc[4];
#pragma unroll
  for (int t = 0; t < 4; ++t) acc[t] = {};

  const float* arow = att + (size_t)row * Dd;
#pragma unroll 2
  for (int k0 = 0; k0 < Dd; k0 += 4) {
    v2f a;
    a.x = arow[k0 + khi];
    a.y = arow[k0 + khi + 1];
    __builtin_prefetch(arow + k0 + 64, 0, 3);
    v2f bw[4];
#pragma unroll
    for (int t = 0; t < 4; ++t) {
      const int n = nb + t * 16 + l16;
      const float* w = wo_w + (size_t)n * Dd + k0 + khi;
      bw[t].x = w[0]; bw[t].y = w[1];
    }
#pragma unroll
    for (int t = 0; t < 4; ++t) acc[t] = WMMA_F32(a, bw[t], acc[t]);
  }

#pragma unroll
  for (int t = 0; t < 4; ++t) {
    const int n = nb + t * 16 + l16;
    const float bias = wo_b[n];
#pragma unroll
    for (int e = 0; e < 8; ++e) {
      const int mr = m0 + e + ((lane >= 16) ? 8 : 0);
      out[(size_t)mr * Dd + n] = acc[t][e] + bias;
    }
  }
}

// ---------------------------------------------------------------------------
extern "C" void kernel_launch(void* const* d_in, const int* in_sizes, int n_in,
                              void* d_out, int out_size, void* d_ws, size_t ws_size,
                              hipStream_t stream) {
  const float* x    = (const float*)d_in[0];
  const float* wq_w = (const float*)d_in[1];
  const float* wq_b = (const float*)d_in[2];
  // d_in[3], d_in[4] = wk_w, wk_b : dead in the no-cache reference path.
  const float* wv_w = (const float*)d_in[5];
  const float* wv_b = (const float*)d_in[6];
  const float* wo_w = (const float*)d_in[7];
  const float* wo_b = (const float*)d_in[8];
  float* out = (float*)d_out;

  const size_t headElems = (size_t)Bb * Hh * Ss * HD;  // 2M floats
  float* xq  = (float*)d_ws;
  float* xv  = xq + headElems;
  float* att = xv + headElems;

  qv_proj_rope_kernel<<<Mrows / 16, 128, 0, stream>>>(x, wq_w, wq_b, wv_w, wv_b, xq, xv);
  attn_kernel<<<Bb * Hh, 512, 0, stream>>>(xq, xv, att);
  out_proj_kernel<<<Mrows / 16, 128, 0, stream>>>(att, wo_w, wo_b, out);
}